// ntm_5454608466466
// MI455X (gfx1250) — compile-verified
//
#include <hip/hip_runtime.h>
#include <hip/hip_bf16.h>
#include <math.h>

#define NROWS 65536
#define MCOLS 512
#define OWLEN 1542
#define ORLEN 518
#define TS    132   // LDS tile stride (128 + 4) -> conflict-free WMMA A reads

typedef __attribute__((ext_vector_type(2))) float v2f;
typedef __attribute__((ext_vector_type(8))) float v8f;

__device__ __forceinline__ float softplusf(float x) {
  return (x > 20.f) ? x : log1pf(expf(x));
}

__device__ __forceinline__ void grid_barrier(unsigned* cnt) {
  __syncthreads();
  if (threadIdx.x == 0) {
    __threadfence();
    unsigned my   = atomicAdd(cnt, 1u) + 1u;
    unsigned nb   = gridDim.x;
    unsigned need = ((my + nb - 1u) / nb) * nb;
    while (*((volatile unsigned*)cnt) < need) { __builtin_amdgcn_s_sleep(1); }
    __threadfence();
  }
  __syncthreads();
}

// ---------------- init: w_w = w_r = 1/N, reset barrier counter ----------------
__global__ void init_kernel(float* __restrict__ w_w, float* __restrict__ w_r,
                            unsigned* __restrict__ cnt) {
  int i = blockIdx.x * 1024 + threadIdx.x;   // grid 64 x 1024 = 65536
  const float v = 1.0f / (float)NROWS;
  w_w[i] = v;
  w_r[i] = v;
  if (i == 0) cnt[0] = 0u;
}

// ---------------- controller: cont = x @ Wc.T + bc (4x128) ----------------
__global__ void cont_kernel(const float* __restrict__ x, const float* __restrict__ Wc,
                            const float* __restrict__ bc, float* __restrict__ cont) {
  int j = threadIdx.x;            // 512 threads
  int t = j >> 7, c = j & 127;
  float acc = bc[c];
  const float* xr = x + t * 128;
  const float* wr = Wc + c * 128;
  for (int i = 0; i < 128; ++i) acc += xr[i] * wr[i];
  cont[j] = acc;
}

// ---------------- ow = cont @ Ww.T + bw ; orr = cont @ Wr.T + br ----------------
__global__ __launch_bounds__(256) void ctrl_out_kernel(
    const float* __restrict__ cont,
    const float* __restrict__ Ww, const float* __restrict__ bw,
    const float* __restrict__ Wr, const float* __restrict__ br,
    float* __restrict__ owAll, float* __restrict__ orrAll) {
  int gid = blockIdx.x * 256 + threadIdx.x;
  if (gid < 4 * OWLEN) {
    int t = gid / OWLEN, o = gid % OWLEN;
    float acc = bw[o];
    const float* c = cont + t * 128;
    const float* w = Ww + (size_t)o * 128;
    for (int i = 0; i < 128; ++i) acc += c[i] * w[i];
    owAll[gid] = acc;
  } else {
    int g2 = gid - 4 * OWLEN;
    if (g2 < 4 * ORLEN) {
      int t = g2 / ORLEN, o = g2 % ORLEN;
      float acc = br[o];
      const float* c = cont + t * 128;
      const float* w = Wr + (size_t)o * 128;
      for (int i = 0; i < 128; ++i) acc += c[i] * w[i];
      orrAll[g2] = acc;
    }
  }
}

// --------- per-head scalar params: beta,g,s0..2,gamma,||k||  (8 blocks) ---------
__global__ __launch_bounds__(256) void prep_params_kernel(
    const float* __restrict__ owAll, const float* __restrict__ orrAll,
    float* __restrict__ params) {
  int b = blockIdx.x;             // 0..7 : (t, which) = (b>>1, b&1); which 0=write 1=read
  int t = b >> 1, which = b & 1;
  const float* raw = which ? (orrAll + t * ORLEN) : (owAll + t * OWLEN);
  __shared__ float sd[256];
  int tid = threadIdx.x;
  float s = 0.f;
  for (int i = tid; i < MCOLS; i += 256) { float v = raw[i]; s += v * v; }
  sd[tid] = s; __syncthreads();
  for (int sft = 128; sft > 0; sft >>= 1) { if (tid < sft) sd[tid] += sd[tid + sft]; __syncthreads(); }
  if (tid == 0) {
    float knorm = sqrtf(sd[0]);
    float beta  = softplusf(raw[MCOLS]);
    float g     = 1.f / (1.f + expf(-raw[MCOLS + 1]));
    float sr0 = raw[MCOLS + 2], sr1 = raw[MCOLS + 3], sr2 = raw[MCOLS + 4];
    float mx = fmaxf(sr0, fmaxf(sr1, sr2));
    float e0 = expf(sr0 - mx), e1 = expf(sr1 - mx), e2 = expf(sr2 - mx);
    float es = e0 + e1 + e2;
    float gamma = 1.f + softplusf(raw[MCOLS + 5]);
    float* P = params + b * 8;
    P[0] = beta; P[1] = g; P[2] = e0 / es; P[3] = e1 / es; P[4] = e2 / es;
    P[5] = gamma; P[6] = knorm; P[7] = 0.f;
  }
}

// ---------------- esig[t][j] = sigmoid(e) ----------------
__global__ void esig_kernel(const float* __restrict__ owAll, float* __restrict__ esig) {
  int t = blockIdx.x, j = threadIdx.x;          // 4 x 512
  float v = owAll[t * OWLEN + MCOLS + 6 + j];
  esig[t * MCOLS + j] = 1.f / (1.f + expf(-v));
}

// ======= fused memory update + WMMA dot products + row norms =======
// grid 1024 x 128 threads (4 waves); one wave owns 16 rows.
// doUpdate==0: pure dot/norm pass on memIn (no writes).
__global__ __launch_bounds__(128) void update_dot_kernel(
    const float* __restrict__ memIn, float* __restrict__ memOut,
    const float* __restrict__ ww, const float* __restrict__ esig,
    const float* __restrict__ aAdd,
    const float* __restrict__ kA, const float* __restrict__ kB,
    float* __restrict__ dotA, float* __restrict__ dotB,
    float* __restrict__ norm2, int doUpdate) {
  __shared__ float tileAll[4 * 16 * TS];
  __shared__ float ldsKA[MCOLS];
  __shared__ float ldsKB[MCOLS];

  const int tid  = threadIdx.x;
  const int wave = tid >> 5;
  const int lane = tid & 31;
  const int n    = lane & 15;
  const int half = lane >> 4;
  const int rowBase = (blockIdx.x * 4 + wave) * 16;
  float* tile = tileAll + wave * 16 * TS;

  for (int i = tid; i < MCOLS; i += 128) { ldsKA[i] = kA[i]; ldsKB[i] = kB[i]; }
  __syncthreads();

  float wwr[16];
  if (doUpdate) {
#pragma unroll
    for (int r = 0; r < 16; ++r) wwr[r] = ww[rowBase + r];
  }

  float normAcc[16];
#pragma unroll
  for (int r = 0; r < 16; ++r) normAcc[r] = 0.f;

  v8f acc;
#pragma unroll
  for (int m = 0; m < 8; ++m) acc[m] = 0.f;

  for (int chunk = 0; chunk < 4; ++chunk) {
    const int cbase = chunk * 128;
    const int col   = cbase + lane * 4;
    float4 ev, av;
    if (doUpdate) {
      ev = *(const float4*)(esig + col);
      av = *(const float4*)(aAdd + col);
    }
#pragma unroll
    for (int r = 0; r < 16; ++r) {
      float4 v = *(const float4*)(memIn + (size_t)(rowBase + r) * MCOLS + col);
      if (doUpdate) {
        const float w = wwr[r];
        v.x = v.x * (1.f - w * ev.x) + w * av.x;
        v.y = v.y * (1.f - w * ev.y) + w * av.y;
        v.z = v.z * (1.f - w * ev.z) + w * av.z;
        v.w = v.w * (1.f - w * ev.w) + w * av.w;
        *(float4*)(memOut + (size_t)(rowBase + r) * MCOLS + col) = v;
      }
      float* trow = tile + r * TS + lane * 4;
      trow[0] = v.x; trow[1] = v.y; trow[2] = v.z; trow[3] = v.w;
      normAcc[r] += v.x * v.x + v.y * v.y + v.z * v.z + v.w * v.w;
    }
    __syncthreads();
    // D(16x16) += A(16x4) x B(4x16), K marched 4 at a time; B cols 0/1 = kA/kB.
#pragma unroll
    for (int kk = 0; kk < 32; ++kk) {
      const int c = kk * 4 + half * 2;
      v2f a, b;
      a.x = tile[n * TS + c];
      a.y = tile[n * TS + c + 1];
      b.x = 0.f; b.y = 0.f;
      if (n == 0)      { b.x = ldsKA[cbase + c]; b.y = ldsKA[cbase + c + 1]; }
      else if (n == 1) { b.x = ldsKB[cbase + c]; b.y = ldsKB[cbase + c + 1]; }
      acc = __builtin_amdgcn_wmma_f32_16x16x4_f32(false, a, false, b, (short)0,
                                                  acc, false, false);
    }
    __syncthreads();
  }

  // D column 0 -> dotA, column 1 -> dotB (lanes n / n+16 hold rows 0-7 / 8-15)
#pragma unroll
  for (int m = 0; m < 8; ++m) {
    if (lane == 0)  dotA[rowBase + m]     = acc[m];
    if (lane == 16) dotA[rowBase + 8 + m] = acc[m];
    if (lane == 1)  dotB[rowBase + m]     = acc[m];
    if (lane == 17) dotB[rowBase + 8 + m] = acc[m];
  }

  float myNorm = 0.f;
#pragma unroll
  for (int r = 0; r < 16; ++r) {
    float v = normAcc[r];
    v += __shfl_xor(v, 16, 32);
    v += __shfl_xor(v, 8, 32);
    v += __shfl_xor(v, 4, 32);
    v += __shfl_xor(v, 2, 32);
    v += __shfl_xor(v, 1, 32);
    if (lane == r) myNorm = v;
  }
  if (lane < 16) norm2[rowBase + lane] = myNorm;
}

// ======= full addressing (softmax, gate, shift, sharpen, renorm) =======
// grid MUST be 64 blocks x 256 threads (software grid barrier).
__global__ __launch_bounds__(256) void addr_kernel(
    const float* __restrict__ dot, const float* __restrict__ norm2,
    const float* __restrict__ p, float* __restrict__ w /* prev in, new out */,
    float* __restrict__ tmp, float* __restrict__ wg,
    float* __restrict__ psum, unsigned* __restrict__ cnt) {
  const int tid = threadIdx.x;
  const int stride = 64 * 256;
  const float beta = p[0], g = p[1], s0 = p[2], s1 = p[3], s2 = p[4];
  const float gamma = p[5], knorm = p[6];
  __shared__ float sd[256];

  // P1: logits -> exp, partial sum   (|logit| <= beta, so no max-shift needed)
  float local = 0.f;
  for (int i = blockIdx.x * 256 + tid; i < NROWS; i += stride) {
    float logit = beta * dot[i] / (sqrtf(norm2[i]) * knorm + 1e-16f);
    float e = expf(logit);
    tmp[i] = e;
    local += e;
  }
  sd[tid] = local; __syncthreads();
  for (int s = 128; s > 0; s >>= 1) { if (tid < s) sd[tid] += sd[tid + s]; __syncthreads(); }
  if (tid == 0) psum[blockIdx.x] = sd[0];
  grid_barrier(cnt);

  // P2: S, wg = g*wc + (1-g)*w_prev
  float S = 0.f;
  for (int b = 0; b < 64; ++b) S += psum[b];
  const float invS = 1.f / S;
  for (int i = blockIdx.x * 256 + tid; i < NROWS; i += stride)
    wg[i] = g * tmp[i] * invS + (1.f - g) * w[i];
  grid_barrier(cnt);

  // P3: circular shift conv + pow(gamma), partial sum
  local = 0.f;
  for (int i = blockIdx.x * 256 + tid; i < NROWS; i += stride) {
    float v = s0 * wg[(i + NROWS - 1) & (NROWS - 1)] + s1 * wg[i]
            + s2 * wg[(i + 1) & (NROWS - 1)];
    float pw = powf(v, gamma);
    tmp[i] = pw;
    local += pw;
  }
  sd[tid] = local; __syncthreads();
  for (int s = 128; s > 0; s >>= 1) { if (tid < s) sd[tid] += sd[tid + s]; __syncthreads(); }
  if (tid == 0) psum[blockIdx.x] = sd[0];
  grid_barrier(cnt);

  // P4: renormalize
  float Sw = 0.f;
  for (int b = 0; b < 64; ++b) Sw += psum[b];
  const float invSw = 1.f / (Sw + 1e-16f);
  for (int i = blockIdx.x * 256 + tid; i < NROWS; i += stride)
    w[i] = tmp[i] * invSw;
}

// ---------------- r = w_r @ mem : per-block partials over 256 rows ----------------
__global__ __launch_bounds__(256) void read_pass_kernel(
    const float* __restrict__ mem, const float* __restrict__ wr,
    float* __restrict__ partial) {
  const int tid = threadIdx.x;
  const int rowBase = blockIdx.x * 256;     // grid 256 blocks
  float a0 = 0.f, a1 = 0.f;
  for (int i = 0; i < 256; ++i) {
    const float w = wr[rowBase + i];
    const float* row = mem + (size_t)(rowBase + i) * MCOLS;
    a0 += w * row[tid];
    a1 += w * row[tid + 256];
  }
  partial[blockIdx.x * MCOLS + tid]       = a0;
  partial[blockIdx.x * MCOLS + tid + 256] = a1;
}

__global__ void read_reduce_kernel(const float* __restrict__ partial,
                                   float* __restrict__ r) {
  int c = threadIdx.x;                      // 512 threads
  float a = 0.f;
  for (int b = 0; b < 256; ++b) a += partial[b * MCOLS + c];
  r[c] = a;
}

// ---------------- final MLP: out = W2 @ relu(W1 @ [c;r] + b1) + b2 ----------------
__global__ __launch_bounds__(256) void mlp_kernel(
    const float* __restrict__ c_t, const float* __restrict__ r,
    const float* __restrict__ W1, const float* __restrict__ b1,
    const float* __restrict__ W2, const float* __restrict__ b2,
    float* __restrict__ out) {
  __shared__ float inb[640];
  __shared__ float h[200];
  int tid = threadIdx.x;
  for (int i = tid; i < 640; i += 256) inb[i] = (i < 128) ? c_t[i] : r[i - 128];
  __syncthreads();
  for (int j = tid; j < 200; j += 256) {
    float acc = b1[j];
    const float* w = W1 + (size_t)j * 640;
    for (int i = 0; i < 640; ++i) acc += w[i] * inb[i];
    h[j] = fmaxf(acc, 0.f);
  }
  __syncthreads();
  if (tid < 10) {
    float acc = b2[tid];
    const float* w = W2 + tid * 200;
    for (int i = 0; i < 200; ++i) acc += w[i] * h[i];
    out[tid] = acc;
  }
}

extern "C" void kernel_launch(void* const* d_in, const int* in_sizes, int n_in,
                              void* d_out, int out_size, void* d_ws, size_t ws_size,
                              hipStream_t stream) {
  const float* x   = (const float*)d_in[0];
  const float* mem = (const float*)d_in[1];
  const float* Wc  = (const float*)d_in[2];
  const float* bc  = (const float*)d_in[3];
  const float* Wr  = (const float*)d_in[4];
  const float* br  = (const float*)d_in[5];
  const float* Ww  = (const float*)d_in[6];
  const float* bw  = (const float*)d_in[7];
  const float* W1  = (const float*)d_in[8];
  const float* b1  = (const float*)d_in[9];
  const float* W2  = (const float*)d_in[10];
  const float* b2  = (const float*)d_in[11];
  float* out = (float*)d_out;

  float* ws = (float*)d_ws;
  size_t off = 0;
  auto alloc = [&](size_t nf) { float* p = ws + off; off += nf; return p; };
  float*    memWs  = alloc((size_t)NROWS * MCOLS);   // 134 MB working copy
  float*    cont   = alloc(512);
  float*    owAll  = alloc(4 * OWLEN);
  float*    orrAll = alloc(4 * ORLEN);
  float*    params = alloc(64);
  float*    esig   = alloc(4 * MCOLS);
  float*    dW     = alloc(NROWS);
  float*    dR     = alloc(NROWS);
  float*    norm2  = alloc(NROWS);
  float*    w_w    = alloc(NROWS);
  float*    w_r    = alloc(NROWS);
  float*    tmp    = alloc(NROWS);
  float*    wgb    = alloc(NROWS);
  float*    psum   = alloc(64);
  unsigned* barCnt = (unsigned*)alloc(16);
  float*    rpart  = alloc(256 * MCOLS);
  float*    rvec   = alloc(4 * MCOLS);

  init_kernel<<<64, 1024, 0, stream>>>(w_w, w_r, barCnt);
  cont_kernel<<<1, 512, 0, stream>>>(x, Wc, bc, cont);
  ctrl_out_kernel<<<(4 * OWLEN + 4 * ORLEN + 255) / 256, 256, 0, stream>>>(
      cont, Ww, bw, Wr, br, owAll, orrAll);
  prep_params_kernel<<<8, 256, 0, stream>>>(owAll, orrAll, params);
  esig_kernel<<<4, 512, 0, stream>>>(owAll, esig);

  // pre-pass: dot with k_w(0) + norms on the (unmodified) input memory
  update_dot_kernel<<<1024, 128, 0, stream>>>(
      mem, memWs, nullptr, nullptr, nullptr,
      owAll /*k_w0*/, owAll, dW, dR /*scratch*/, norm2, 0);

  for (int t = 0; t < 4; ++t) {
    const float* owT  = owAll + t * OWLEN;
    const float* orrT = orrAll + t * ORLEN;
    // write-head addressing (pre-write mem state)
    addr_kernel<<<64, 256, 0, stream>>>(dW, norm2, params + (2 * t + 0) * 8,
                                        w_w, tmp, wgb, psum, barCnt);
    // fused: erase/add update + dots vs k_r(t) & k_w(t+1) + new norms
    const float* kB = (t < 3) ? (owAll + (t + 1) * OWLEN) : orrT;
    update_dot_kernel<<<1024, 128, 0, stream>>>(
        (t == 0) ? mem : memWs, memWs, w_w, esig + t * MCOLS,
        owT + 2 * MCOLS + 6 /*a*/, orrT /*k_r*/, kB, dR, dW, norm2, 1);
    // read-head addressing (post-write mem state)
    addr_kernel<<<64, 256, 0, stream>>>(dR, norm2, params + (2 * t + 1) * 8,
                                        w_r, tmp, wgb, psum, barCnt);
    // r = w_r @ mem
    read_pass_kernel<<<256, 256, 0, stream>>>(memWs, w_r, rpart);
    read_reduce_kernel<<<1, 512, 0, stream>>>(rpart, rvec + t * MCOLS);
    // output head
    mlp_kernel<<<1, 256, 0, stream>>>(cont + t * 128, rvec + t * MCOLS,
                                      W1, b1, W2, b2, out + t * 10);
  }
}